// EKVConv2d_27144193310932
// MI455X (gfx1250) — compile-verified
//
#include <hip/hip_runtime.h>
#include <stdint.h>

#ifndef __has_builtin
#define __has_builtin(x) 0
#endif

// Tensor Data Mover path (gfx1250): guarded so host pass / older toolchains fall back.
#if __has_builtin(__builtin_amdgcn_tensor_load_to_lds) && __has_builtin(__builtin_amdgcn_s_wait_tensorcnt)
#define EKV_USE_TDM 1
#else
#define EKV_USE_TDM 0
#endif

namespace {
constexpr int BATCH = 4;
constexpr int CIN   = 16;
constexpr int HH    = 64;
constexpr int WW    = 64;
constexpr int OUTC  = 32;
constexpr int KTOT  = CIN * 3 * 3;   // 144
constexpr int OPB   = 8;             // output channels per block
constexpr int THN   = OPB * KTOT;    // 1152 theta elements staged per block

constexpr float kAlpha = 0.0005625f;
constexpr float kInv   = 1.0f / (1.5f * 0.026f);   // 1/(n*Vt)
constexpr float kC2    = 0.1f * (1.0f / (1.5f * 0.026f)); // VD/(n*Vt)
}

#if EKV_USE_TDM
typedef __attribute__((ext_vector_type(4))) unsigned tdm_u4;
typedef __attribute__((ext_vector_type(8))) int      tdm_i8;
typedef __attribute__((ext_vector_type(4))) int      tdm_i4;
#endif

__device__ __forceinline__ float softplus_c(float v) {
  // reference: log1p(exp(clip(v, -30, 30))); compiler forms v_med3 + v_exp + v_log
  v = fminf(fmaxf(v, -30.0f), 30.0f);
  return __logf(1.0f + __expf(v));
}

__global__ __launch_bounds__(256)
void ekv_conv2d_kernel(const float* __restrict__ x,
                       const float* __restrict__ theta,
                       float* __restrict__ out) {
  __shared__ float thinv[THN];         // theta * inv for this block's 8 out-channels
  __shared__ float xs[3 * CIN * WW];   // rows h-1,h,h+1 for all 16 input channels

  const int tid   = threadIdx.x;
  const int blk   = blockIdx.x;          // b*256 + h*4 + og4
  const int og4   = blk & 3;
  const int h     = (blk >> 2) & 63;
  const int b     = blk >> 8;
  const int obase = og4 * OPB;

  // ---------------- stage theta slice (8 x 144 f32, contiguous) ----------------
#if EKV_USE_TDM
  if (tid == 0) {
    unsigned long long ga = (unsigned long long)(uintptr_t)(theta + obase * KTOT);
    unsigned ldsOff = (unsigned)(uintptr_t)(&thinv[0]);  // low 32 bits == LDS byte offset
    // D# group 0: count=1 (valid user desc), lds_addr, global_addr[56:0], type=2
    tdm_u4 g0 = { 1u,
                  ldsOff,
                  (unsigned)ga,
                  (unsigned)((ga >> 32) & 0x01FFFFFFu) | (2u << 30) };
    // D# group 1: data_size=4B; tensor_dim0=1152; tensor_dim1=1; tile_dim0=1152;
    //             tile_dim1=1; tile_dim2=0; dim0_stride=1152; dim1_stride=1152
    tdm_i8 g1 = { (int)(2u << 16),
                  (int)((unsigned)THN << 16),   // tensor_dim0[15:0] @ bits 63:48
                  (int)(1u << 16),              // tensor_dim0[31:16]=0 | tensor_dim1[15:0]=1
                  (int)((unsigned)THN << 16),   // tensor_dim1 hi=0 | tile_dim0
                  1,                            // tile_dim1=1, tile_dim2=0
                  THN,                          // tensor_dim0_stride[31:0]
                  (int)((unsigned)THN << 16),   // stride0 hi=0 | tensor_dim1_stride[15:0]
                  0 };
    tdm_i4 gz4 = { 0, 0, 0, 0 };                // groups 2/3 unused (<=2D tensor)
    tdm_i8 gz8 = { 0, 0, 0, 0, 0, 0, 0, 0 };    // extra group (6-arg builtin form), unused
    __builtin_amdgcn_tensor_load_to_lds(g0, g1, gz4, gz4, gz8, 0);
    __builtin_amdgcn_s_wait_tensorcnt(0);
  }
#else
  for (int i = tid; i < THN; i += 256) thinv[i] = theta[obase * KTOT + i];
#endif

  // ------------- stage 3 input rows, zero-filled outside the image -------------
  // 3 rows * 16 chans * 64 floats = 768 float4 chunks; 3 per thread, coalesced
  for (int i = tid; i < 3 * CIN * (WW / 4); i += 256) {
    const int lane16 = i & 15;
    const int rc     = i >> 4;     // 0..47 = r*16 + c
    const int c      = rc & 15;
    const int r      = rc >> 4;
    const int row    = h - 1 + r;
    float4* dst = reinterpret_cast<float4*>(&xs[rc * WW + lane16 * 4]);
    if ((unsigned)row < (unsigned)HH) {
      const float4* src = reinterpret_cast<const float4*>(
          &x[(((b * CIN + c) * HH + row) * WW) + lane16 * 4]);
      *dst = *src;
    } else {
      *dst = make_float4(0.f, 0.f, 0.f, 0.f);
    }
  }
  __syncthreads();

  // hoist the 1/(n*Vt) scale into the theta table (saves a mul per pair)
  for (int i = tid; i < THN; i += 256) thinv[i] *= kInv;
  __syncthreads();

  // ---------------------------------- compute ----------------------------------
  const int wave = tid >> 5;                 // 8 wave32s
  const int lane = tid & 31;
  const int w    = ((wave & 1) << 5) + lane; // contiguous 32-wide w per wave
  const int o0   = (wave >> 1) * 2;          // wave owns 2 output channels

  float acc0 = 0.f, acc1 = 0.f;

  #pragma unroll 1
  for (int c = 0; c < CIN; ++c) {
    #pragma unroll 1
    for (int dh = 0; dh < 3; ++dh) {
      const float* xrow = &xs[(dh * CIN + c) * WW];
      const int kb = c * 9 + dh * 3;
      #pragma unroll
      for (int dw = 0; dw < 3; ++dw) {
        const int col  = w + dw - 1;
        // out-of-range column == reference's zero padding
        const float vg = ((unsigned)col < (unsigned)WW) ? xrow[col] : 0.0f;
        const float vi = vg * kInv;          // shared across both o's
        const int   kk = kb + dw;
        {
          const float th = thinv[o0 * KTOT + kk];          // wave-uniform -> LDS broadcast
          const float a1 = vi - th;
          const float s1 = softplus_c(a1);
          const float s2 = softplus_c(a1 - kC2);
          acc0 = fmaf(s1, s1, acc0);
          acc0 = fmaf(-s2, s2, acc0);
        }
        {
          const float th = thinv[(o0 + 1) * KTOT + kk];
          const float a1 = vi - th;
          const float s1 = softplus_c(a1);
          const float s2 = softplus_c(a1 - kC2);
          acc1 = fmaf(s1, s1, acc1);
          acc1 = fmaf(-s2, s2, acc1);
        }
      }
    }
  }

  const int oA = obase + o0;
  out[(((b * OUTC + oA    ) * HH + h) * WW) + w] = kAlpha * acc0;
  out[(((b * OUTC + oA + 1) * HH + h) * WW) + w] = kAlpha * acc1;
}

extern "C" void kernel_launch(void* const* d_in, const int* in_sizes, int n_in,
                              void* d_out, int out_size, void* d_ws, size_t ws_size,
                              hipStream_t stream) {
  (void)in_sizes; (void)n_in; (void)d_ws; (void)ws_size; (void)out_size;
  const float* x     = (const float*)d_in[0];   // (4,16,64,64) f32
  const float* theta = (const float*)d_in[1];   // (32,144) f32
  float* out         = (float*)d_out;           // (4,32,64,64) f32

  dim3 grid(BATCH * HH * (OUTC / OPB));  // 1024 blocks
  dim3 block(256);                       // 8 wave32s
  hipLaunchKernelGGL(ekv_conv2d_kernel, grid, block, 0, stream, x, theta, out);
}